// Irnn_6047313953546
// MI455X (gfx1250) — compile-verified
//
#include <hip/hip_runtime.h>

// ---------------------------------------------------------------------------
// IRNN 4-direction scan for MI455X (gfx1250, wave32, WMMA + TDM).
//   h_i = x_i + h_{i-1} @ W  computed as Ht_new = Wt * Ht_prev + Xt via
//   V_WMMA_F32_16X16X4_F32 (A = Wt chunk, resident; B = carry; C = X tile).
// Memory-bound (~400 MB @ 23.3 TB/s ~ 17us) -> fp32 kept, coalesced global
// traffic; horizontal scans staged through LDS by the Tensor Data Mover
// (tensor_load_to_lds, TENSORcnt double-buffered).
// ---------------------------------------------------------------------------

typedef float v2f __attribute__((ext_vector_type(2)));
typedef float v8f __attribute__((ext_vector_type(8)));
typedef unsigned int u32x4 __attribute__((ext_vector_type(4)));
typedef unsigned int u32x8 __attribute__((ext_vector_type(8)));

#if defined(__HIP_DEVICE_COMPILE__)
#if !__has_builtin(__builtin_amdgcn_wmma_f32_16x16x4_f32)
#error "missing __builtin_amdgcn_wmma_f32_16x16x4_f32 on this toolchain"
#endif
#endif

#define IMG_S 256
#define IMG_SS 65536   // 256*256
#define NCH 32

// Swap low/high 16 lanes of a wave32 (ds_swizzle SWAPX16: xor=0x10,and=0x1f)
__device__ __forceinline__ float swap16(float x) {
  return __int_as_float(__builtin_amdgcn_ds_swizzle(__float_as_int(x), 0x401F));
}

__device__ __forceinline__ v8f wmma4(v2f a, v2f b, v8f c) {
  // (neg_a, A, neg_b, B, c_mod, C, reuse_a, reuse_b)
  return __builtin_amdgcn_wmma_f32_16x16x4_f32(false, a, false, b, (short)0, c,
                                               false, false);
}

// D = chain of 8 K-chunks: D = A7*B7 + (... (A0*B0 + C))
__device__ __forceinline__ v8f chain8(const v2f* a, const v2f* b, v8f c) {
#pragma unroll
  for (int kb = 0; kb < 8; ++kb) c = wmma4(a[kb], b[kb], c);
  return c;
}

// Build B-operand chunk KB (in-channels 4*KB..4*KB+3) from a D tile.
// Rows stay in lane%16 in both layouts; only a half-wave swap + cndmask.
// NOTE: swizzles are hoisted OUT of the selects so they execute with full
// EXEC and the select lowers to v_cndmask (no saveexec branches).
template <int KB>
__device__ __forceinline__ v2f mk_b(const v8f& d, bool hi) {
  constexpr int base = (KB & 1) * 4;   // reg of lowest channel in chunk
  constexpr int s = (KB >> 1) & 1;     // source half holding these channels
  v2f r;
  if constexpr (s == 0) {
    const float sx = swap16(d[base + 2]);
    const float sy = swap16(d[base + 3]);
    r.x = hi ? sx : d[base + 0];
    r.y = hi ? sy : d[base + 1];
  } else {
    const float sx = swap16(d[base + 0]);
    const float sy = swap16(d[base + 1]);
    r.x = hi ? d[base + 2] : sx;
    r.y = hi ? d[base + 3] : sy;
  }
  return r;
}

__device__ __forceinline__ void relayout(const v8f& d0, const v8f& d1, v2f* b,
                                         bool hi) {
  b[0] = mk_b<0>(d0, hi); b[1] = mk_b<1>(d0, hi);
  b[2] = mk_b<2>(d0, hi); b[3] = mk_b<3>(d0, hi);
  b[4] = mk_b<4>(d1, hi); b[5] = mk_b<5>(d1, hi);
  b[6] = mk_b<6>(d1, hi); b[7] = mk_b<7>(d1, hi);
}

// A-operand = Wt chunks: A[kb] VGPR v, lane l: M(out)=l%16(+16*tile),
// K(in) = 4*kb + v + 2*(l>>4); value = W[in*32 + out]. Loaded once.
__device__ __forceinline__ void load_w(const float* __restrict__ W, int ll,
                                       int hl, v2f* a0, v2f* a1) {
#pragma unroll
  for (int kb = 0; kb < 8; ++kb) {
    const int i0 = 4 * kb + 2 * hl;  // v=0 in-channel
    const int i1 = i0 + 1;           // v=1 in-channel
    a0[kb].x = W[i0 * 32 + ll];
    a0[kb].y = W[i1 * 32 + ll];
    a1[kb].x = W[i0 * 32 + 16 + ll];
    a1[kb].y = W[i1 * 32 + 16 + ll];
  }
}

// ---------------------------------------------------------------------------
// TDM: one tensor_load_to_lds moves a 3D tile (16 cols x 16 rows x 32 ch,
// f32, strides 1/256/65536 elems) into LDS, padding 1 dword per 16 written
// (-> pitch-17 rows). Tracked by TENSORcnt. CDNA5 ISA-8.
// ---------------------------------------------------------------------------
__device__ __forceinline__ void tdm_load_tile(const float* gptr,
                                              unsigned lds_byte_off) {
  const unsigned long long ga = (unsigned long long)(size_t)gptr;
  u32x4 g0;
  g0[0] = 1u;                                   // count=1, user mode
  g0[1] = lds_byte_off;                         // LDS byte address
  g0[2] = (unsigned)(ga & 0xffffffffu);         // global_addr[31:0]
  g0[3] = (unsigned)((ga >> 32) & 0x01ffffffu)  // global_addr[56:32]
          | (2u << 30);                         // type=2 ("image")
  u32x8 g1;
  g1[0] = (2u << 16)      // data_size = 4B
          | (1u << 20)    // pad_enable
          | (3u << 22)    // pad_interval: 16 dwords
          | (0u << 25);   // pad_amount: 1 dword
  g1[1] = (256u << 16);   // tensor_dim0[15:0] in [31:16]
  g1[2] = (256u << 16);   // tensor_dim0[31:16]=0, tensor_dim1[15:0]=256
  g1[3] = (16u << 16);    // tensor_dim1[31:16]=0, tile_dim0=16
  g1[4] = 16u | (32u << 16);  // tile_dim1=16, tile_dim2=32
  g1[5] = 256u;           // tensor_dim0_stride[31:0] = 256 (row stride)
  g1[6] = 0u;             // stride0[47:32]=0, tensor_dim1_stride[15:0]=0
  g1[7] = 1u;             // tensor_dim1_stride[47:16]=1 -> stride = 65536 (ch)
  u32x4 g2;
  g2[0] = 32u;            // tensor_dim2 = 32
  g2[1] = 0u;             // tensor_dim3 = 0
  g2[2] = 0u;             // tensor_dim2_stride = 0 (unused)
  g2[3] = 0u;             // tile_dim3 = 0
  u32x4 g3;
  g3[0] = 0u; g3[1] = 0u; g3[2] = 0u; g3[3] = 0u;
  asm volatile("tensor_load_to_lds %0, %1, %2, %3"
               :
               : "s"(g0), "s"(g1), "s"(g2), "s"(g3)
               : "memory");
}

// ---------------------------------------------------------------------------
// Vertical scans (axis=2, out channels 64..95 fwd / 96..127 bwd).
// Wave = (n, 16-column strip, dir). Per row-step every global access is two
// contiguous 64B runs -> natively coalesced, no LDS.
// ---------------------------------------------------------------------------
__global__ __launch_bounds__(32) void irnn_vert(const float* __restrict__ x,
                                                const float* __restrict__ W,
                                                float* __restrict__ out) {
  const int bx = blockIdx.x;
  const int dir = bx >> 7;
  const int n = (bx >> 4) & 7;
  const int c0 = (bx & 15) * 16;
  const int lane = threadIdx.x;
  const int ll = lane & 15, hl = lane >> 4;
  const bool hi = hl != 0;

  v2f a0[8], a1[8];
  load_w(W, ll, hl, a0, a1);

  v2f b[8];
#pragma unroll
  for (int k = 0; k < 8; ++k) { b[k].x = 0.f; b[k].y = 0.f; }

  const float* xb = x + (size_t)n * NCH * IMG_SS + c0 + ll;
  float* ob = out + ((size_t)n * 128 + 64 + 32 * dir) * IMG_SS + c0 + ll;

  for (int step = 0; step < IMG_S; ++step) {
    const int row = dir ? (IMG_S - 1 - step) : step;
    const int roff = row * IMG_S;
    v8f c0v, c1v;
#pragma unroll
    for (int p = 0; p < 8; ++p) {
      const int ch = p + 8 * hl;
      c0v[p] = xb[(size_t)ch * IMG_SS + roff];
      c1v[p] = xb[(size_t)(ch + 16) * IMG_SS + roff];
    }
    const v8f d0 = chain8(a0, b, c0v);
    const v8f d1 = chain8(a1, b, c1v);
#pragma unroll
    for (int p = 0; p < 8; ++p) {
      const int ch = p + 8 * hl;
      ob[(size_t)ch * IMG_SS + roff] = d0[p];
      ob[(size_t)(ch + 16) * IMG_SS + roff] = d1[p];
    }
    relayout(d0, d1, b, hi);
  }
}

// ---------------------------------------------------------------------------
// Horizontal scans (axis=3, out channels 0..31 fwd / 32..63 bwd).
// Wave = (n, 16-row strip, dir). X chunks DMA'd into LDS by the TDM
// (double-buffered on TENSORcnt); results staged in a pitch-20 out-buffer
// and flushed as coalesced float4 stores (64B runs).
// ---------------------------------------------------------------------------
#define XWORDS (512 * 17)  // one x staging buffer: 512 (ch,row) pairs, pitch 17
#define OPITCH 20          // dwords per (oc,row) pair in out staging
#define CHUNK 16           // columns per staging chunk

__global__ __launch_bounds__(32) void irnn_horiz(const float* __restrict__ x,
                                                 const float* __restrict__ W,
                                                 float* __restrict__ out) {
  extern __shared__ float lds[];
  float* xb0 = lds;               // XWORDS dwords
  float* xb1 = lds + XWORDS;      // XWORDS dwords
  float* obuf = lds + 2 * XWORDS; // 512 * OPITCH dwords

  const int bx = blockIdx.x;
  const int dir = bx >> 7;
  const int n = (bx >> 4) & 7;
  const int r0 = (bx & 15) * 16;
  const int lane = threadIdx.x;
  const int ll = lane & 15, hl = lane >> 4;
  const bool hi = hl != 0;

  v2f a0[8], a1[8];
  load_w(W, ll, hl, a0, a1);

  v2f b[8];
#pragma unroll
  for (int k = 0; k < 8; ++k) { b[k].x = 0.f; b[k].y = 0.f; }

  // Tile base for this (n, row-strip); per chunk add its column offset.
  const float* tile0 = x + ((size_t)(n * NCH) * IMG_S + r0) * IMG_S;
  const unsigned ldsbase = (unsigned)__builtin_amdgcn_groupstaticsize();
  const unsigned xoff[2] = {ldsbase, ldsbase + (unsigned)(XWORDS * 4)};

  // Prologue: DMA first chunk.
  {
    const int ck0 = dir ? 15 : 0;
    tdm_load_tile(tile0 + ck0 * CHUNK, xoff[0]);
  }

  for (int m = 0; m < 16; ++m) {
    const int chunk = dir ? (15 - m) : m;
    const int cb = chunk * CHUNK;

    // Wait for chunk m (the only outstanding TDM op), then launch m+1 into
    // the other buffer so the DMA overlaps this chunk's 16 scan steps.
    __builtin_amdgcn_s_wait_tensorcnt(0);
    if (m + 1 < 16) {
      const int cknext = dir ? (15 - (m + 1)) : (m + 1);
      tdm_load_tile(tile0 + cknext * CHUNK, xoff[(m + 1) & 1]);
    }
    const float* xbuf = (m & 1) ? xb1 : xb0;

    for (int j = 0; j < CHUNK; ++j) {
      const int cl = dir ? (CHUNK - 1 - j) : j;
      v8f c0v, c1v;
#pragma unroll
      for (int p = 0; p < 8; ++p) {  // pitch 17: conflict-free within halves
        c0v[p] = xbuf[((p + 8 * hl) * 16 + ll) * 17 + cl];
        c1v[p] = xbuf[((16 + p + 8 * hl) * 16 + ll) * 17 + cl];
      }
      const v8f d0 = chain8(a0, b, c0v);
      const v8f d1 = chain8(a1, b, c1v);
#pragma unroll
      for (int p = 0; p < 8; ++p) {
        obuf[((p + 8 * hl) * 16 + ll) * OPITCH + cl] = d0[p];
        obuf[((16 + p + 8 * hl) * 16 + ll) * OPITCH + cl] = d1[p];
      }
      relayout(d0, d1, b, hi);
    }

    // Flush out staging -> global: per (oc,row) a 64B contiguous run.
#pragma unroll 4
    for (int it = 0; it < 64; ++it) {
      const int pair = it * 8 + (lane >> 2);
      const int oc = pair >> 4;
      const int r = pair & 15;
      const int cg = lane & 3;
      const float4 v = *(const float4*)(obuf + pair * OPITCH + 4 * cg);
      *(float4*)(out + (((size_t)(n * 128 + 32 * dir + oc)) * IMG_S +
                        (r0 + r)) * IMG_S + cb + 4 * cg) = v;
    }
  }
}

// ---------------------------------------------------------------------------
extern "C" void kernel_launch(void* const* d_in, const int* in_sizes, int n_in,
                              void* d_out, int out_size, void* d_ws,
                              size_t ws_size, hipStream_t stream) {
  const float* x = (const float*)d_in[0];  // (8,32,256,256) f32
  const float* w = (const float*)d_in[1];  // (32,32) f32
  float* out = (float*)d_out;              // (8,128,256,256) f32

  constexpr size_t kLds = (size_t)(2 * XWORDS + 512 * OPITCH) * 4;  // 110592B

  // 256 one-wave workgroups each: (8 n) x (16 strips) x (2 dirs)
  irnn_vert<<<dim3(256), dim3(32), 0, stream>>>(x, w, out);
  irnn_horiz<<<dim3(256), dim3(32), kLds, stream>>>(x, w, out);
}